// ReverbModel_67508295958534
// MI455X (gfx1250) — compile-verified
//
#include <hip/hip_runtime.h>
#include <hip/hip_bf16.h>

typedef __attribute__((ext_vector_type(2))) float v2f;
typedef __attribute__((ext_vector_type(8))) float v8f;

#define LANES   4410            // DELAY
#define CHUNKS  6000            // n / DELAY
#define TILE    16              // chunks advanced per WMMA chain
#define TPB     15              // output tiles per chunk-block (240 chunks)
#define GBLK    25              // chunk-blocks: TPB*GBLK*TILE == CHUNKS
#define WARM    2               // warm-up tiles (32 chunks): 2^-32 carry error << fp32 ulp
#define NSTRIP  ((LANES + 15) / 16)   // 276 lane-strips of 16 columns
#define NWAVES  (NSTRIP * GBLK)       // 6900 independent waves
#define TSTEP   (TILE * LANES)        // 70560 elements per tile step

__device__ __forceinline__ float pow2i(int e) {
    // exact power of two, e in [-126, 127]
    return __int_as_float((e + 127) << 23);
}

struct B4 { v2f b[4]; };

// Load one 16x16 X tile into the B-operand layout for four 16x16x4 f32 WMMAs.
// xi already encodes (chunk_base + krow)*LANES + clamped column, so every
// intra-tile offset is a compile-time constant that folds into the VMEM
// instruction's 24-bit immediate offset (saddr + voffset + ioffset form).
__device__ __forceinline__ B4 load_btile(const float* __restrict__ x, unsigned xi) {
    B4 r;
#pragma unroll
    for (int s = 0; s < 4; ++s) {
        r.b[s].x = x[xi + (unsigned)((4 * s)     * LANES)];   // K = 4s + {0|1}
        r.b[s].y = x[xi + (unsigned)((4 * s + 2) * LANES)];   // K = 4s + {2|3}
    }
    return r;
}

// y[c][l] = x[c][l] + 0.5 * y[c-1][l], carry seeded 0 at c = 0.
// Per 16x16 tile:  Y = A * X + (2^-(M+1) ⊗ carry),  A lower-triangular powers of 0.5.
// Columns are independent in Y = A*X, so padded columns (strip 275) carry
// garbage harmlessly: loads are clamped in-bounds, stores are masked.
__global__ __launch_bounds__(256) void comb_scan_wmma(const float* __restrict__ x,
                                                      float* __restrict__ y) {
    const int wave = blockIdx.x * (blockDim.x >> 5) + (threadIdx.x >> 5);
    if (wave >= NWAVES) return;                 // uniform per-wave exit
    const int lane  = threadIdx.x & 31;
    const int strip = wave % NSTRIP;
    const int gblk  = wave / NSTRIP;

    const int  col   = strip * 16 + (lane & 15);
    const bool colok = col < LANES;
    const int  colc  = colok ? col : (LANES - 1);   // clamped load column

    // ---- A matrix (16x4 per K-slice), ISA layout: lanes0-15 -> K = 4s+{0,1},
    //      lanes16-31 -> K = 4s+{2,3};  A[m][j] = (m>=j) ? 2^(j-m) : 0
    const int m  = lane & 15;
    const int kb = (lane < 16) ? 0 : 2;
    v2f a[4];
#pragma unroll
    for (int s = 0; s < 4; ++s) {
        const int jx = 4 * s + kb, jy = jx + 1;
        a[s].x = (m >= jx) ? pow2i(jx - m) : 0.0f;
        a[s].y = (m >= jy) ? pow2i(jy - m) : 0.0f;
    }

    // ---- carry weights 2^-(M+1) for the 8 D-rows this lane owns
    const int mofs = (lane < 16) ? 0 : 8;       // C/D: VGPR r -> M = r + mofs
    float pw[8];
#pragma unroll
    for (int r = 0; r < 8; ++r) pw[r] = pow2i(-(r + mofs + 1));

    const int warm  = (gblk == 0) ? 0 : WARM;
    const int cbase = gblk * (TPB * TILE) - warm * TILE;    // first (warm-up) chunk
    const int krow  = (lane < 16) ? 0 : 1;  // B: VGPR0 holds K rows 0/1, VGPR1 rows 2/3

    // 32-bit element indices (buffer < 2^31 elements); advanced by TSTEP per tile
    unsigned xi = (unsigned)(cbase + krow) * LANES + (unsigned)colc;
    unsigned yi = (unsigned)(cbase + mofs) * LANES + (unsigned)col;

    float carry = 0.0f;
    B4 cur = load_btile(x, xi);                 // preload first tile

    for (int t = -warm; t < TPB; ++t) {
        // ---- software pipeline: issue next tile's loads before this tile's WMMAs
        const unsigned xin = xi + ((t + 1 < TPB) ? (unsigned)TSTEP : 0u);
        B4 nxt = load_btile(x, xin);
        if (t + 2 < TPB)                        // stream-ahead -> global_prefetch_b8
            __builtin_prefetch(&x[xi + 2u * TSTEP], 0, 1);

        // ---- accumulator seeded with carry contribution C[M][n] = 2^-(M+1)*carry_n
        v8f acc;
#pragma unroll
        for (int r = 0; r < 8; ++r) acc[r] = pw[r] * carry;

        // ---- 16x16x16 product as four chained 16x16x4 f32 WMMAs
#pragma unroll
        for (int s = 0; s < 4; ++s) {
            acc = __builtin_amdgcn_wmma_f32_16x16x4_f32(
                false, a[s], false, cur.b[s], (short)0, acc, false, false);
        }

        // ---- store outputs (skip warm-up tiles; mask padded columns)
        if (t >= 0 && colok) {
#pragma unroll
            for (int r = 0; r < 8; ++r)
                y[yi + (unsigned)(r * LANES)] = acc[r];
        }

        // ---- new carry_n = Y[M=15][n] = acc[7] in lane 16+n; broadcast to both halves
        const int src = (16 + (lane & 15)) << 2;
        carry = __int_as_float(__builtin_amdgcn_ds_bpermute(src, __float_as_int(acc[7])));

        cur = nxt;
        xi  = xin;
        yi += (unsigned)TSTEP;
    }
}

extern "C" void kernel_launch(void* const* d_in, const int* in_sizes, int n_in,
                              void* d_out, int out_size, void* d_ws, size_t ws_size,
                              hipStream_t stream) {
    (void)in_sizes; (void)n_in; (void)d_ws; (void)ws_size; (void)out_size;
    const float* x = (const float*)d_in[0];
    float*       y = (float*)d_out;

    const int wavesPerBlock = 256 / 32;
    const int grid = (NWAVES + wavesPerBlock - 1) / wavesPerBlock;  // 863
    comb_scan_wmma<<<grid, 256, 0, stream>>>(x, y);
}